// TripletLoss_35974646071812
// MI455X (gfx1250) — compile-verified
//
#include <hip/hip_runtime.h>
#include <hip/hip_bf16.h>
#include <math.h>

typedef __attribute__((ext_vector_type(2))) float v2f;
typedef __attribute__((ext_vector_type(8))) float v8f;

#define TL_MARGIN 0.3f
#define D_DIM 256          // embedding dim (reference fixes D=256)
#define KT_TILES 64        // D / 4 k-steps per 16x16 output tile
#define ROW_PAD 260        // 256 + 4 dword pad -> conflict-free ds_load_b64 pattern
#define USE_ASYNC_STAGE 1  // 1: global_load_async_to_lds_b128; 0: plain ld/st staging

// Order-preserving float<->uint encoding so we can use integer atomicMin/Max.
__device__ __forceinline__ unsigned encF(float f) {
    unsigned u = __float_as_uint(f);
    return (u & 0x80000000u) ? ~u : (u | 0x80000000u);
}
__device__ __forceinline__ float decF(unsigned e) {
    unsigned b = (e & 0x80000000u) ? (e & 0x7FFFFFFFu) : ~e;
    return __uint_as_float(b);
}

__global__ void tl_init_kernel(unsigned* __restrict__ pos_enc,
                               unsigned* __restrict__ neg_enc, int B) {
    int i = blockIdx.x * blockDim.x + threadIdx.x;
    if (i < B) {
        pos_enc[i] = 0xFFFFFFFFu;  // min-identity; sentinel = "no positive seen"
        neg_enc[i] = 0u;           // max-identity; sentinel = "no negative seen"
    }
}

// Block: 256 threads = 8 waves, 128 anchor rows. Each wave keeps a 16x256 f32
// A tile resident in VGPRs. The 16x256 B tile for the current column tile is
// staged ONCE per block into double-buffered LDS (async global->LDS copies),
// cutting L2 read traffic 8x. Per tile: 64 chained v_wmma_f32_16x16x4_f32
// fed from LDS, then a fused hard-mining epilogue in sim-space.
__global__ void __launch_bounds__(256, 2)
tl_sim_mine_kernel(const float* __restrict__ emb,
                   const int* __restrict__ labels,
                   unsigned* __restrict__ pos_enc,
                   unsigned* __restrict__ neg_enc,
                   int tilesPerChunk) {
    __shared__ float Bbuf[2][16][ROW_PAD];  // ~33 KB

    const int tid   = threadIdx.x;
    const int lane  = tid & 31;
    const int wave  = tid >> 5;        // 0..7
    const int m     = lane & 15;       // row-within-tile / col-within-tile
    const int hi    = lane >> 4;       // 0 or 1 (K / M half selector)
    const int rbase = blockIdx.x * 128 + wave * 16;
    const int cbase = blockIdx.y * tilesPerChunk * 16;

    // Staging role: thread copies 4x16B chunks of row `srow` of the B tile.
    const int srow = tid >> 4;         // 0..15
    const int slp  = tid & 15;         // 16 threads per row

    // --- Load resident A tile: lane supplies E[rbase+m][4*kt + 2*hi .. +1] ---
    const float* arow = emb + (size_t)(rbase + m) * D_DIM + 2 * hi;
    v2f Areg[KT_TILES];
#pragma unroll
    for (int kt = 0; kt < KT_TILES; ++kt) {
        const float2 t = *reinterpret_cast<const float2*>(arow + kt * 4);
        Areg[kt] = v2f{t.x, t.y};
    }

    // Row labels for the 8 M-rows this lane's C registers cover.
    int lrow[8];
#pragma unroll
    for (int j = 0; j < 8; ++j) lrow[j] = labels[rbase + j + 8 * hi];

    float posmin[8], negmax[8];
#pragma unroll
    for (int j = 0; j < 8; ++j) { posmin[j] = INFINITY; negmax[j] = -INFINITY; }

    // ---- stage one 16x256 B tile (16 KB) into Bbuf[buf] cooperatively ----
    auto stage = [&](int buf, int col0) {
        const float* gsrc = emb + (size_t)(col0 + srow) * D_DIM + slp * 4;
#pragma unroll
        for (int p = 0; p < 4; ++p) {
#if USE_ASYNC_STAGE
            unsigned long long ga =
                (unsigned long long)(const void*)(gsrc + p * 64);
            // AS3 pointers are LDS byte offsets; flat = aperture<<32 | offset.
            unsigned ldsoff = (unsigned)(unsigned long long)(const void*)
                                  &Bbuf[buf][srow][(p * 16 + slp) * 4];
            asm volatile("global_load_async_to_lds_b128 %0, %1, off"
                         :: "v"(ldsoff), "v"(ga)
                         : "memory");
#else
            const float4 v = *reinterpret_cast<const float4*>(gsrc + p * 64);
            *reinterpret_cast<float4*>(&Bbuf[buf][srow][(p * 16 + slp) * 4]) = v;
#endif
        }
    };
    auto stage_drain = [&]() {
#if USE_ASYNC_STAGE
        asm volatile("s_wait_asynccnt 0x0" ::: "memory");
#endif
    };

    stage(0, cbase);
    stage_drain();
    __syncthreads();

    for (int ct = 0; ct < tilesPerChunk; ++ct) {
        const int cur  = ct & 1;
        const int col0 = cbase + ct * 16;

        if (ct + 1 < tilesPerChunk) stage(cur ^ 1, cbase + (ct + 1) * 16);

        // B fragment source in LDS: row m, dwords 4*kt + 2*hi (+1).
        const float* bbase = &Bbuf[cur][m][2 * hi];

        v8f acc0 = {};
        v8f acc1 = {};
#pragma unroll
        for (int g = 0; g < 8; ++g) {
            v2f Bt[8];
#pragma unroll
            for (int t = 0; t < 8; ++t) {
                const float2 bv =
                    *reinterpret_cast<const float2*>(bbase + (g * 8 + t) * 4);
                Bt[t] = v2f{bv.x, bv.y};
            }
#pragma unroll
            for (int t = 0; t < 8; ++t) {
                if (t & 1)
                    acc1 = __builtin_amdgcn_wmma_f32_16x16x4_f32(
                        false, Areg[g * 8 + t], false, Bt[t],
                        (short)0, acc1, false, false);
                else
                    acc0 = __builtin_amdgcn_wmma_f32_16x16x4_f32(
                        false, Areg[g * 8 + t], false, Bt[t],
                        (short)0, acc0, false, false);
            }
        }

        // C/D layout: VGPR j holds sim[M][N], M = j + 8*hi, N = m.
        const int ccol = col0 + m;
        const int lcol = labels[ccol];
#pragma unroll
        for (int j = 0; j < 8; ++j) {
            const float s    = acc0[j] + acc1[j];
            const int   irow = rbase + j + 8 * hi;
            const bool  same = (lrow[j] == lcol);
            const bool  diag = (irow == ccol);
            if (same && !diag) posmin[j] = fminf(posmin[j], s);  // hardest pos
            if (!same)         negmax[j] = fmaxf(negmax[j], s);  // hardest neg
        }

        stage_drain();       // our async writes for next tile are in LDS
        __syncthreads();     // everyone done reading cur + done staging next
    }

    // Butterfly reduce across the 16-lane half that shares each M-row set.
#pragma unroll
    for (int j = 0; j < 8; ++j) {
        float p = posmin[j], n = negmax[j];
#pragma unroll
        for (int mask = 8; mask >= 1; mask >>= 1) {
            p = fminf(p, __shfl_xor(p, mask, 32));
            n = fmaxf(n, __shfl_xor(n, mask, 32));
        }
        if (m == 0) {  // lanes 0 and 16 publish rows rbase+j and rbase+8+j
            const int irow = rbase + j + 8 * hi;
            if (p < INFINITY)  atomicMin(&pos_enc[irow], encF(p));
            if (n > -INFINITY) atomicMax(&neg_enc[irow], encF(n));
        }
    }
}

__global__ void __launch_bounds__(256)
tl_finalize_kernel(const unsigned* __restrict__ pos_enc,
                   const unsigned* __restrict__ neg_enc,
                   float* __restrict__ out, int B) {
    __shared__ float ssum[256];
    __shared__ int   scnt[256];
    float sum = 0.0f;
    int   cnt = 0;
    for (int i = threadIdx.x; i < B; i += 256) {
        const unsigned up = pos_enc[i], un = neg_enc[i];
        if (up != 0xFFFFFFFFu && un != 0u) {  // valid: >=1 positive and negative
            const float pmin = decF(up), nmax = decF(un);
            const float dap = sqrtf(fmaxf(2.0f - 2.0f * pmin, 0.0f) + 1e-12f);
            const float dan = sqrtf(fmaxf(2.0f - 2.0f * nmax, 0.0f) + 1e-12f);
            sum += fmaxf(dap - dan + TL_MARGIN, 0.0f);
            cnt += 1;
        }
    }
    ssum[threadIdx.x] = sum;
    scnt[threadIdx.x] = cnt;
    __syncthreads();
    for (int s = 128; s > 0; s >>= 1) {
        if ((int)threadIdx.x < s) {
            ssum[threadIdx.x] += ssum[threadIdx.x + s];
            scnt[threadIdx.x] += scnt[threadIdx.x + s];
        }
        __syncthreads();
    }
    if (threadIdx.x == 0) {
        const float nv = (float)scnt[0];
        out[0] = (nv > 0.0f) ? (ssum[0] / fmaxf(nv, 1.0f)) : 0.0f;
    }
}

extern "C" void kernel_launch(void* const* d_in, const int* in_sizes, int n_in,
                              void* d_out, int out_size, void* d_ws, size_t ws_size,
                              hipStream_t stream) {
    const float* emb    = (const float*)d_in[0];
    const int*   labels = (const int*)d_in[1];
    float*       out    = (float*)d_out;

    const int B = in_sizes[1];  // 8192 anchors

    unsigned* pos_enc = (unsigned*)d_ws;
    unsigned* neg_enc = pos_enc + B;

    tl_init_kernel<<<(B + 255) / 256, 256, 0, stream>>>(pos_enc, neg_enc, B);

    // 8 waves x 16 rows = 128 rows per block; split columns into 16 chunks.
    const int nColTiles = B / 16;            // 512
    int chunks = 16;
    while (chunks > 1 && (nColTiles % chunks)) chunks >>= 1;
    const int tilesPerChunk = nColTiles / chunks;  // 32 for B=8192

    dim3 grid(B / 128, chunks);
    tl_sim_mine_kernel<<<grid, 256, 0, stream>>>(emb, labels, pos_enc, neg_enc,
                                                 tilesPerChunk);

    tl_finalize_kernel<<<1, 256, 0, stream>>>(pos_enc, neg_enc, out, B);
}